// DiffEmbedding1234_85332410237041
// MI455X (gfx1250) — compile-verified
//
#include <hip/hip_runtime.h>
#include <hip/hip_bf16.h>
#include <stdint.h>

typedef __attribute__((ext_vector_type(2))) float v2f;
typedef __attribute__((ext_vector_type(8))) float v8f;

#define B_  32
#define C_  32
#define T_  2048
#define D_  512

// -------- CDNA5 async global->LDS copy (ASYNCcnt path), with sync fallback ---
#if defined(__gfx1250__) && __has_builtin(__builtin_amdgcn_global_load_async_to_lds_b128) && __has_builtin(__builtin_amdgcn_s_wait_asynccnt)
#define USE_ASYNC_LDS 1
#else
#define USE_ASYNC_LDS 0
#endif

#if USE_ASYNC_LDS
typedef int v4i_ __attribute__((vector_size(16)));
typedef __attribute__((address_space(1))) v4i_* g_v4i_p;  // global ptr to int4
typedef __attribute__((address_space(3))) v4i_* l_v4i_p;  // LDS ptr to int4
#endif

__device__ __forceinline__ void copy16_to_lds(const float* __restrict__ g,
                                              float* __restrict__ l) {
#if USE_ASYNC_LDS
    // Generic->AS casts via integer round-trip (low 32 bits of a generic LDS
    // address are the LDS offset per the aperture truncation rule).
    __builtin_amdgcn_global_load_async_to_lds_b128(
        (g_v4i_p)(uintptr_t)g,
        (l_v4i_p)(unsigned int)(uintptr_t)l, 0, 0);
#else
    *(float4*)l = *(const float4*)g;
#endif
}

__device__ __forceinline__ void wait_async_copies() {
#if USE_ASYNC_LDS
    __builtin_amdgcn_s_wait_asynccnt(0);
#endif
}

// ---------------------------------------------------------------------------
// Kernel 1: x[B,C,T] -> diff along T -> GEMM1 (C=32 -> D=512) + bias
// Each wave computes a 16(t) x 64(d) tile with V_WMMA_F32_16X16X4_F32.
// grid = (T/128, B, D/64), block = 256 (8 waves)
// ---------------------------------------------------------------------------
__global__ void emb_kernel(const float* __restrict__ x,
                           const float* __restrict__ Wve,
                           const float* __restrict__ bve,
                           float* __restrict__ emb) {
    const int wave  = threadIdx.x >> 5;   // 0..7 (wave32)
    const int lane  = threadIdx.x & 31;
    const int mrow  = lane & 15;          // M (or N) index within tile
    const int khalf = lane >> 4;          // 0: K pair {0,1}, 1: K pair {2,3}

    const int b  = blockIdx.y;
    const int t0 = blockIdx.x * 128 + wave * 16;
    const int n0 = blockIdx.z * 64;

    v8f acc0 = {}, acc1 = {}, acc2 = {}, acc3 = {};

    const int t = t0 + mrow;
    // Branchless first-difference: at t==0 read x[t] twice -> diff = 0.
    const int poff = (t > 0) ? -1 : 0;

#pragma unroll
    for (int k0 = 0; k0 < C_; k0 += 4) {
        const int c0 = k0 + khalf * 2;
        const float* xp = x + ((size_t)b * C_ + c0) * T_ + t;
        v2f a;
        a.x = xp[0]  - xp[poff];
        a.y = xp[T_] - xp[T_ + poff];

        // B fragments: B[k][n] = Wve[n][k]  (Wve is [D, C] row-major)
#pragma unroll
        for (int j = 0; j < 4; ++j) {
            const int n = n0 + j * 16 + mrow;
            const float* wp = Wve + (size_t)n * C_ + c0;
            v2f bf;
            bf.x = wp[0];
            bf.y = wp[1];
            v8f cin = (j == 0) ? acc0 : (j == 1) ? acc1 : (j == 2) ? acc2 : acc3;
            v8f d = __builtin_amdgcn_wmma_f32_16x16x4_f32(
                false, a, false, bf, (short)0, cin, false, false);
            if (j == 0) acc0 = d; else if (j == 1) acc1 = d;
            else if (j == 2) acc2 = d; else acc3 = d;
        }
    }

    // Epilogue: add b_ve (commutes exactly through the EWMA) and store.
#pragma unroll
    for (int j = 0; j < 4; ++j) {
        const int n = n0 + j * 16 + mrow;
        const float bias = bve[n];
        v8f a = (j == 0) ? acc0 : (j == 1) ? acc1 : (j == 2) ? acc2 : acc3;
#pragma unroll
        for (int r = 0; r < 8; ++r) {
            const int m = khalf * 8 + r;  // D-matrix layout: lanes>=16 hold M=8+r
            emb[((size_t)b * T_ + (t0 + m)) * D_ + n] = a[r] + bias;
        }
    }
}

// ---------------------------------------------------------------------------
// Kernel 2: bidirectional EWMA over T. One thread per (b, d) column.
// Forward scan -> fbuf; backward scan combines and overwrites emb in place
// (emb becomes x_smooth). Consecutive threads hit consecutive d -> coalesced.
// ---------------------------------------------------------------------------
__global__ void scan_kernel(float* __restrict__ emb,   // in: x_emb, out: x_smooth
                            float* __restrict__ fbuf) { // scratch for forward scan
    const int gid = blockIdx.x * blockDim.x + threadIdx.x;  // 0 .. B*D-1
    const int b = gid / D_;
    const int d = gid % D_;
    const size_t base = (size_t)b * T_ * D_ + d;

    const float alpha = 0.1f, beta = 0.9f;

    // forward: f[0] = y[0]; f[t] = a*y[t] + (1-a)*f[t-1]
    float carry = emb[base];
    fbuf[base] = carry;
#pragma unroll 4
    for (int t = 1; t < T_; ++t) {
        const size_t idx = base + (size_t)t * D_;
        carry = fmaf(beta, carry, alpha * emb[idx]);
        fbuf[idx] = carry;
    }

    // backward: g[T-1] = y[T-1]; g[t] = a*y[t] + (1-a)*g[t+1]
    {
        const size_t idx = base + (size_t)(T_ - 1) * D_;
        carry = emb[idx];
        emb[idx] = 0.5f * (fbuf[idx] + carry);
    }
#pragma unroll 4
    for (int t = T_ - 2; t >= 0; --t) {
        const size_t idx = base + (size_t)t * D_;
        const float y = emb[idx];
        carry = fmaf(beta, carry, alpha * y);
        emb[idx] = 0.5f * (fbuf[idx] + carry);
    }
}

// ---------------------------------------------------------------------------
// Kernel 3: out = smooth[65536,512] @ W_lin^T + b_lin  via f32 WMMA.
// Block tile: 128M x 64N, K in 32-wide chunks, double-buffered LDS staging
// with GLOBAL_LOAD_ASYNC_TO_LDS_B128 (ASYNCcnt) overlapping the WMMA chain.
// LDS row stride 36 floats (144 B): rows 16B-aligned for b128, and the 16x4
// f32 fragment reads are bank-conflict-free (36*m mod 64 distinct, khalf
// groups hit disjoint bank quads).
// grid = (M/128, N/64), block = 256 (8 waves)
// ---------------------------------------------------------------------------
#define KCHUNK 32
#define LDSTR  36   // 32 data + 4 pad floats
#define NCHNK  (D_ / KCHUNK)

__global__ void out_gemm_kernel(const float* __restrict__ S,
                                const float* __restrict__ Wlin,
                                const float* __restrict__ blin,
                                float* __restrict__ out) {
    __shared__ float Alds[2][128 * LDSTR];
    __shared__ float Blds[2][64 * LDSTR];

    const int tid   = threadIdx.x;
    const int wave  = tid >> 5;
    const int lane  = tid & 31;
    const int mrow  = lane & 15;
    const int khalf = lane >> 4;

    const int m0 = blockIdx.x * 128;
    const int n0 = blockIdx.y * 64;

    v8f acc0 = {}, acc1 = {}, acc2 = {}, acc3 = {};

    auto stage = [&](int buf, int k0) {
        // A panel: 128 rows x 32 k = 1024 float4 tasks, 4 per thread
#pragma unroll
        for (int it = 0; it < 4; ++it) {
            const int task = tid + it * 256;
            const int m  = task >> 3;
            const int kq = (task & 7) * 4;
            copy16_to_lds(S + (size_t)(m0 + m) * D_ + k0 + kq,
                          &Alds[buf][m * LDSTR + kq]);
        }
        // B panel: B[k][n] = Wlin[n][k]; 64 rows x 32 k, 2 per thread
#pragma unroll
        for (int it = 0; it < 2; ++it) {
            const int task = tid + it * 256;
            const int n  = task >> 3;
            const int kq = (task & 7) * 4;
            copy16_to_lds(Wlin + (size_t)(n0 + n) * D_ + k0 + kq,
                          &Blds[buf][n * LDSTR + kq]);
        }
    };

    stage(0, 0);

    for (int kc = 0; kc < NCHNK; ++kc) {
        wait_async_copies();     // own async stage(kc) complete
        __syncthreads();         // everyone's stage(kc) complete, buf[kc^1] free

        if (kc + 1 < NCHNK) stage((kc + 1) & 1, (kc + 1) * KCHUNK);

        const float* Ab = Alds[kc & 1];
        const float* Bb = Blds[kc & 1];

#pragma unroll
        for (int k = 0; k < KCHUNK; k += 4) {
            const int kk = k + khalf * 2;
            const v2f a = *(const v2f*)(&Ab[(wave * 16 + mrow) * LDSTR + kk]);
#pragma unroll
            for (int j = 0; j < 4; ++j) {
                const v2f bf = *(const v2f*)(&Bb[(j * 16 + mrow) * LDSTR + kk]);
                v8f cin = (j == 0) ? acc0 : (j == 1) ? acc1 : (j == 2) ? acc2 : acc3;
                v8f d = __builtin_amdgcn_wmma_f32_16x16x4_f32(
                    false, a, false, bf, (short)0, cin, false, false);
                if (j == 0) acc0 = d; else if (j == 1) acc1 = d;
                else if (j == 2) acc2 = d; else acc3 = d;
            }
        }
    }

    // Epilogue: add bias, store 16x64 tile per wave.
#pragma unroll
    for (int j = 0; j < 4; ++j) {
        const int n = n0 + j * 16 + mrow;
        const float bias = blin[n];
        v8f a = (j == 0) ? acc0 : (j == 1) ? acc1 : (j == 2) ? acc2 : acc3;
#pragma unroll
        for (int r = 0; r < 8; ++r) {
            const int m = m0 + wave * 16 + khalf * 8 + r;
            out[(size_t)m * D_ + n] = a[r] + bias;
        }
    }
}

// ---------------------------------------------------------------------------
extern "C" void kernel_launch(void* const* d_in, const int* in_sizes, int n_in,
                              void* d_out, int out_size, void* d_ws, size_t ws_size,
                              hipStream_t stream) {
    const float* x    = (const float*)d_in[0];  // [B, C, T]
    const float* Wve  = (const float*)d_in[1];  // [D, C]
    const float* bve  = (const float*)d_in[2];  // [D]
    const float* Wlin = (const float*)d_in[3];  // [D, D]
    const float* blin = (const float*)d_in[4];  // [D]
    float* out = (float*)d_out;                 // [B, T, D]

    const size_t elems = (size_t)B_ * T_ * D_;  // 33,554,432
    float* buf_emb = (float*)d_ws;              // 128 MB: x_emb -> x_smooth (in place)
    float* buf_fwd = buf_emb + elems;           // 128 MB: forward-scan scratch

    // 1) diff + value-embedding GEMM + bias
    emb_kernel<<<dim3(T_ / 128, B_, D_ / 64), 256, 0, stream>>>(x, Wve, bve, buf_emb);

    // 2) bidirectional EWMA (buf_emb becomes x_smooth)
    scan_kernel<<<(B_ * D_) / 256, 256, 0, stream>>>(buf_emb, buf_fwd);

    // 3) output projection GEMM + bias
    out_gemm_kernel<<<dim3((B_ * T_) / 128, D_ / 64), 256, 0, stream>>>(
        buf_emb, Wlin, blin, out);
}